// SoftPattern_2207613190520
// MI455X (gfx1250) — compile-verified
//
#include <hip/hip_runtime.h>
#include <math.h>

typedef __attribute__((ext_vector_type(2))) float v2f;
typedef __attribute__((ext_vector_type(8))) float v8f;
typedef __attribute__((ext_vector_type(4))) int   v4i;

#define L_TOTAL   65536
#define DIMK      1024
#define KSTEPS    (DIMK / 4)        // 256 WMMA k-steps of 4
#define ROWS_PER_BLOCK 128          // 8 waves * 16 rows
#define CHUNK     256
#define NCHUNK    (L_TOTAL / CHUNK) // 256
#define BPACK_N   (KSTEPS * 32)     // 8192 v2f = 64 KB

#if __has_builtin(__builtin_amdgcn_global_load_async_to_lds_b128)
#define HAVE_ASYNC_LDS 1
#endif

typedef __attribute__((address_space(1))) v4i g_v4i;   // global int4
typedef __attribute__((address_space(3))) v4i l_v4i;   // LDS int4

// weight-column -> offset into w (P,P,DIM): diag cols 0..7, superdiag cols 8..14
__device__ __forceinline__ size_t wcol_off(int col) {
    return (col < 8) ? (size_t)col * 9 * DIMK
                     : ((size_t)(col - 8) * 9 + 1) * DIMK;
}

// ---- kernel 0: biases = log||w_col|| AND pack B fragments to global --------
__global__ void sp_prep_kernel(const float* __restrict__ w,
                               float* __restrict__ bias,
                               v2f*   __restrict__ bpackg) {
    __shared__ float acc[16];
    int t = threadIdx.x;
    if (t < 16) acc[t] = 0.f;
    __syncthreads();
    if (t < 240) {                      // 15 cols x 16 segments of 64 elems
        int col = t >> 4, seg = t & 15;
        const float* v = w + wcol_off(col) + seg * 64;
        float p = 0.f;
#pragma unroll 8
        for (int i = 0; i < 64; ++i) { float x = v[i]; p += x * x; }
        atomicAdd(&acc[col], p);        // ds_add_f32
    }
    __syncthreads();
    if (t < 15)  bias[t]  = 0.5f * __logf(acc[t]);
    if (t == 15) bias[15] = 0.f;

    // Pack B in WMMA 16x16x4 B-layout: entry (n, lane):
    //   N = lane&15 (weight col), K pair = 4n + 2*(lane>>4) + {0,1}
    for (int idx = t; idx < BPACK_N; idx += 256) {
        int n   = idx >> 5;
        int ln  = idx & 31;
        int col = ln & 15;
        int kb  = 4 * n + ((ln >> 4) << 1);
        v2f v; v.x = 0.f; v.y = 0.f;
        if (col < 15) {
            const float* wv = w + wcol_off(col);
            v.x = wv[kb];
            v.y = wv[kb + 1];
        }
        bpackg[idx] = v;
    }
}

// ---- kernel 1: gathered GEMM + sigmoid via WMMA f32 ------------------------
// out ab[l][0..7]=a, ab[l][8..14]=b, ab[l][15]=pad
__global__ void sp_gemm_kernel(const int*   __restrict__ doc,
                               const float* __restrict__ emb,
                               const v2f*   __restrict__ bpackg,
                               const float* __restrict__ bias,
                               float* __restrict__ ab) {
    __shared__ v2f   bpack[BPACK_N];   // 64 KB (of 320 KB/WGP)
    __shared__ float sbias[16];

    const int tid = threadIdx.x;

#ifdef HAVE_ASYNC_LDS
    // gfx1250 async copy: global -> LDS without VGPR round trip (ASYNCcnt)
    for (int i = tid; i < BPACK_N / 2; i += 256) {   // 16B per op
        __builtin_amdgcn_global_load_async_to_lds_b128(
            (g_v4i*)(bpackg + 2 * i),
            (l_v4i*)(bpack + 2 * i),
            0, 0);
    }
    if (tid < 16) sbias[tid] = bias[tid];
#if __has_builtin(__builtin_amdgcn_s_wait_asynccnt)
    __builtin_amdgcn_s_wait_asynccnt(0);
#else
    asm volatile("s_wait_asynccnt 0x0" ::: "memory");
#endif
    __syncthreads();
#else
    for (int i = tid; i < BPACK_N; i += 256) bpack[i] = bpackg[i];
    if (tid < 16) sbias[tid] = bias[tid];
    __syncthreads();
#endif

    const int wave = tid >> 5;
    const int lane = tid & 31;
    const int rowbase = blockIdx.x * ROWS_PER_BLOCK + wave * 16;

    // A fragment row: lane&15 = M within tile; lane>>4 selects K pair
    const int    docid = doc[rowbase + (lane & 15)];
    const float* rowp  = emb + (size_t)docid * DIMK + ((lane >> 4) << 1);

    // stream the rest of the 4KB row through L2 -> global_prefetch_b8
    __builtin_prefetch(rowp + 256, 0, 0);
    __builtin_prefetch(rowp + 512, 0, 0);
    __builtin_prefetch(rowp + 768, 0, 0);

    v8f c = {};
#pragma unroll 8
    for (int n = 0; n < KSTEPS; ++n) {
        v2f av = *(const v2f*)(rowp + 4 * n);   // 8B per lane, contiguous
        v2f bv = bpack[n * 32 + lane];
        c = __builtin_amdgcn_wmma_f32_16x16x4_f32(
                false, av, false, bv, (short)0, c, false, false);
    }

    const float lb = sbias[lane & 15];
    const int mbase = (lane >> 4) << 3;   // C/D layout: VGPR r -> M=r or 8+r
#pragma unroll
    for (int r = 0; r < 8; ++r) {
        int row = rowbase + mbase + r;
        // sigmoid(dot - log||w||) with fast v_rcp_f32 (no IEEE div chain)
        float sig = __builtin_amdgcn_rcpf(1.f + __expf(lb - c[r]));
        ab[(size_t)row * 16 + (lane & 15)] = sig;
    }
}

// ---- kernel 2: per-chunk composed 9x9 affine -------------------------------
// state s = (h0..h7, score); step: h'=T h + e0, score += h'[7]
__global__ void sp_chunk_kernel(const float* __restrict__ ab,
                                float* __restrict__ aff) {
    int chunk = blockIdx.x * blockDim.x + threadIdx.x;
    if (chunk >= NCHUNK) return;

    float A[9][9], u[9];
#pragma unroll
    for (int i = 0; i < 9; ++i) {
        u[i] = 0.f;
#pragma unroll
        for (int j = 0; j < 9; ++j) A[i][j] = (i == j) ? 1.f : 0.f;
    }

    const float* base = ab + (size_t)chunk * CHUNK * 16;
    for (int s = 0; s < CHUNK; ++s) {
        const float4* r4 = (const float4*)(base + s * 16);
        float4 x0 = r4[0], x1 = r4[1], x2 = r4[2], x3 = r4[3];
        float av[8] = {x0.x, x0.y, x0.z, x0.w, x1.x, x1.y, x1.z, x1.w};
        float bv[7] = {x2.x, x2.y, x2.z, x2.w, x3.x, x3.y, x3.z};

#pragma unroll
        for (int col = 0; col < 9; ++col) {     // A <- T * A
            float t[9];
            t[0] = av[0] * A[0][col];
#pragma unroll
            for (int j = 1; j < 8; ++j)
                t[j] = av[j] * A[j][col] + bv[j - 1] * A[j - 1][col];
            t[8] = A[8][col] + t[7];            // score row: +a7*h7+b6*h6
#pragma unroll
            for (int j = 0; j < 9; ++j) A[j][col] = t[j];
        }
        {                                       // u <- T*u + e0
            float t[9];
            t[0] = av[0] * u[0];
#pragma unroll
            for (int j = 1; j < 8; ++j)
                t[j] = av[j] * u[j] + bv[j - 1] * u[j - 1];
            t[8] = u[8] + t[7];
#pragma unroll
            for (int j = 0; j < 9; ++j) u[j] = t[j];
            u[0] += 1.f;
        }
    }

    float* o = aff + (size_t)chunk * 96;
#pragma unroll
    for (int i = 0; i < 9; ++i)
#pragma unroll
        for (int j = 0; j < 9; ++j) o[i * 9 + j] = A[i][j];
#pragma unroll
    for (int i = 0; i < 9; ++i) o[81 + i] = u[i];
}

// ---- kernel 3: ordered fold of chunk affines (1 wave) ----------------------
__global__ void sp_combine_kernel(const float* __restrict__ aff,
                                  float* __restrict__ out) {
    int lane = threadIdx.x;
    float s = (lane == 0) ? 1.f : 0.f;          // s_init = (start=e0, score=0)
    for (int c = 0; c < NCHUNK; ++c) {
        const float* base = aff + (size_t)c * 96;
        float acc = (lane < 9) ? base[81 + lane] : 0.f;
#pragma unroll
        for (int k = 0; k < 9; ++k) {
            float sk = __shfl(s, k, 32);
            float a  = (lane < 9) ? base[lane * 9 + k] : 0.f;
            acc += a * sk;
        }
        s = acc;
    }
    float score = __shfl(s, 8, 32);
    if (lane == 0) out[0] = score;
}

// ---------------------------------------------------------------------------
extern "C" void kernel_launch(void* const* d_in, const int* in_sizes, int n_in,
                              void* d_out, int out_size, void* d_ws, size_t ws_size,
                              hipStream_t stream) {
    const int*   doc = (const int*)d_in[0];
    const float* emb = (const float*)d_in[1];
    const float* w   = (const float*)d_in[2];
    float* out  = (float*)d_out;

    float* ab   = (float*)d_ws;                    // L*16 f32 = 4 MB
    float* aff  = ab + (size_t)L_TOTAL * 16;       // NCHUNK*96 f32
    float* bias = aff + (size_t)NCHUNK * 96;       // 16 f32
    v2f*   bpackg = (v2f*)(bias + 16);             // 8192 v2f = 64 KB (16B aligned)

    sp_prep_kernel<<<1, 256, 0, stream>>>(w, bias, bpackg);
    sp_gemm_kernel<<<L_TOTAL / ROWS_PER_BLOCK, 256, 0, stream>>>(doc, emb, bpackg, bias, ab);
    sp_chunk_kernel<<<1, 256, 0, stream>>>(ab, aff);
    sp_combine_kernel<<<1, 32, 0, stream>>>(aff, out);
}